// GCN_16862041604211
// MI455X (gfx1250) — compile-verified
//
#include <hip/hip_runtime.h>

// ---------------------------------------------------------------------------
// 3-layer GCN for MI455X (gfx1250, wave32).
//
// Key restructure: segment_sum(h[src] + edge_fea) is split into
//   segment_sum(h[src])  (layer-dependent, L2-resident working set)
// + segment_sum(edge_fea) (layer-INVARIANT, precomputed once -> "eagg")
// so the 307 MB edge_fea stream is read from HBM exactly once instead of 3x.
// edge_fea is loaded with TH=NT so the one-shot stream does not evict the
// node-sized buffers (25.6 MB each) that must stay resident in the 192 MB L2;
// per-layer scatter/gather and the 76.8M f32 atomics are then L2-side ops.
//
// GEMM uses V_WMMA_F32_16X16X4_F32 (full fp32; GEMM FLOPs are negligible).
// ---------------------------------------------------------------------------

typedef float v2f __attribute__((ext_vector_type(2)));
typedef float v4f __attribute__((ext_vector_type(4)));   // clang ext-vector: legal for NT builtins
typedef float v8f __attribute__((ext_vector_type(8)));

#define N_NODES 50000
#define N_EDGES 600000
#define D 128

// ---- inv_sqrt_deg = clip(deg,1)^(-1/2), in place ---------------------------
__global__ void gcn_inv_kernel(float* __restrict__ deg) {
    int i = blockIdx.x * blockDim.x + threadIdx.x;
    if (i < N_NODES) deg[i] = 1.0f / sqrtf(fmaxf(deg[i], 1.0f));
}

// ---- once: eagg[dst[e]] += edge_fea[e], deg[dst[e]] += 1 -------------------
// One wave (32 lanes) per edge; NT loads keep the 307 MB edge_fea stream from
// evicting the L2-resident node buffers. Lane 0 also counts the degree.
__global__ __launch_bounds__(256) void gcn_scatter_ef_kernel(
    const float* __restrict__ ef, const int* __restrict__ dst,
    float* __restrict__ eagg, float* __restrict__ deg) {
    long long tid = (long long)blockIdx.x * blockDim.x + threadIdx.x;
    int e    = (int)(tid >> 5);
    int lane = (int)(tid & 31);
    if (e >= N_EDGES) return;
    int d = dst[e];
    const v4f ev = __builtin_nontemporal_load(
        reinterpret_cast<const v4f*>(ef + (size_t)e * D + lane * 4));
    if (lane == 0) atomicAdd(&deg[d], 1.0f);
    float* ap = eagg + (size_t)d * D + lane * 4;
    atomicAdd(ap + 0, ev.x);
    atomicAdd(ap + 1, ev.y);
    atomicAdd(ap + 2, ev.z);
    atomicAdd(ap + 3, ev.w);
}

// ---- per layer: hagg[dst[e]] += h[src[e]] ----------------------------------
// One wave (32 lanes) per edge; each lane handles 4 contiguous floats.
// h and hagg are L2-resident; regular temporal loads are correct here.
__global__ __launch_bounds__(256) void gcn_scatter_h_kernel(
    const float* __restrict__ h, const int* __restrict__ src,
    const int* __restrict__ dst, float* __restrict__ hagg) {
    long long tid = (long long)blockIdx.x * blockDim.x + threadIdx.x;
    int e    = (int)(tid >> 5);
    int lane = (int)(tid & 31);
    if (e >= N_EDGES) return;
    int s = src[e];
    int d = dst[e];
    const v4f hv = *reinterpret_cast<const v4f*>(h + (size_t)s * D + lane * 4);
    float* ap = hagg + (size_t)d * D + lane * 4;
    atomicAdd(ap + 0, hv.x);
    atomicAdd(ap + 1, hv.y);
    atomicAdd(ap + 2, hv.z);
    atomicAdd(ap + 3, hv.w);
}

// ---- GEMM + epilogue: out = act( ((hagg+eagg) @ W + b) * inv [+ res] ) -----
// Block = 256 threads = 8 waves. Block owns 16 rows (one M tile); wave w owns
// columns [w*16, w*16+16). K accumulated in steps of 4 via WMMA f32 16x16x4.
// A-tile = hagg + eagg summed during the LDS staging load (no copy pass).
// LDS row stride 132 floats: 16B-aligned float4 stores, conflict-free frag
// reads (bank = (4*m + k) % 64 distinct across the 16 lanes of a half-wave).
// LAST=true marks the final layer: output is written NT (never re-read).
template <bool RESIDUAL, bool RELU, bool LAST>
__global__ __launch_bounds__(256) void gcn_gemm_kernel(
    const float* __restrict__ hagg, const float* __restrict__ eagg,
    const float* __restrict__ W, const float* __restrict__ bias,
    const float* __restrict__ inv, const float* __restrict__ residual,
    float* __restrict__ out) {
    __shared__ float As[16][132];

    const int tile    = blockIdx.x;      // M tile index (N_NODES/16 == 3125 exact)
    const int t       = threadIdx.x;
    const int rowBase = tile * 16;

    // Cooperative load of the 16x128 A tile (512 float4s across 256 threads),
    // summing the layer-dependent and layer-invariant aggregates on the fly.
    for (int p = t; p < 512; p += 256) {
        int r = p >> 5;
        int c = (p & 31) * 4;
        const size_t off = (size_t)(rowBase + r) * D + c;
        const v4f hv = *reinterpret_cast<const v4f*>(hagg + off);
        const v4f ev = *reinterpret_cast<const v4f*>(eagg + off);
        *reinterpret_cast<v4f*>(&As[r][c]) = hv + ev;
    }
    __syncthreads();

    const int w       = t >> 5;          // wave id -> N tile
    const int l       = t & 31;          // lane id (wave32)
    const int m       = l & 15;          // A row / B col within tile
    const int hk      = (l >> 4) * 2;    // K sub-offset per half-wave
    const int colBase = w * 16;
    const int col     = colBase + m;

    v8f acc = {};
#pragma unroll 4
    for (int kb = 0; kb < D; kb += 4) {
        const int k = kb + hk;
        v2f a, b;
        a.x = As[m][k];
        a.y = As[m][k + 1];
        b.x = W[(size_t)k * D + col];
        b.y = W[(size_t)(k + 1) * D + col];
        acc = __builtin_amdgcn_wmma_f32_16x16x4_f32(
            /*neg_a=*/false, a, /*neg_b=*/false, b,
            /*c_mod=*/(short)0, acc, /*reuse_a=*/false, /*reuse_b=*/false);
    }

    const float bn = bias[col];
#pragma unroll
    for (int g = 0; g < 8; ++g) {
        const int mrow = g + (l >> 4) * 8;      // C/D layout: VGPR g, halves +8
        const int row  = rowBase + mrow;
        float v = (acc[g] + bn) * inv[row];
        if (RESIDUAL) v += residual[(size_t)row * D + (colBase + m)];
        if (RELU) v = fmaxf(v, 0.0f);
        float* op = out + (size_t)row * D + (colBase + m);
        if (LAST) {
            __builtin_nontemporal_store(v, op);  // final output, never re-read
        } else {
            *op = v;                              // consumed next layer (L2)
        }
    }
}

extern "C" void kernel_launch(void* const* d_in, const int* in_sizes, int n_in,
                              void* d_out, int out_size, void* d_ws, size_t ws_size,
                              hipStream_t stream) {
    const float* node_fea = (const float*)d_in[0];
    const float* edge_fea = (const float*)d_in[1];
    const int*   src      = (const int*)d_in[2];
    const int*   dst      = (const int*)d_in[3];
    const float* W0       = (const float*)d_in[4];
    const float* b0       = (const float*)d_in[5];
    const float* W1       = (const float*)d_in[6];
    const float* b1       = (const float*)d_in[7];
    const float* W2       = (const float*)d_in[8];
    const float* b2       = (const float*)d_in[9];
    float* out = (float*)d_out;

    const size_t ND = (size_t)N_NODES * D;
    float* eagg = (float*)d_ws;          // layer-invariant edge aggregate
    float* hagg = eagg + ND;             // per-layer node aggregate
    float* hA   = hagg + ND;
    float* hB   = hA + ND;
    float* inv  = hB + ND;               // deg, transformed in place

    const int scatterBlocks = (int)(((long long)N_EDGES * 32 + 255) / 256);
    const int gemmBlocks    = N_NODES / 16;   // 3125, exact

    // ---- once per call: edge_fea aggregate + degree, then inv_sqrt ---------
    (void)hipMemsetAsync(inv, 0, N_NODES * sizeof(float), stream);
    (void)hipMemsetAsync(eagg, 0, ND * sizeof(float), stream);
    gcn_scatter_ef_kernel<<<scatterBlocks, 256, 0, stream>>>(edge_fea, dst, eagg, inv);
    gcn_inv_kernel<<<(N_NODES + 255) / 256, 256, 0, stream>>>(inv);

    // ---- layer 0: residual + relu, node_fea -> hA --------------------------
    (void)hipMemsetAsync(hagg, 0, ND * sizeof(float), stream);
    gcn_scatter_h_kernel<<<scatterBlocks, 256, 0, stream>>>(node_fea, src, dst, hagg);
    gcn_gemm_kernel<true, true, false><<<gemmBlocks, 256, 0, stream>>>(
        hagg, eagg, W0, b0, inv, node_fea, hA);

    // ---- layer 1: residual + relu, hA -> hB --------------------------------
    (void)hipMemsetAsync(hagg, 0, ND * sizeof(float), stream);
    gcn_scatter_h_kernel<<<scatterBlocks, 256, 0, stream>>>(hA, src, dst, hagg);
    gcn_gemm_kernel<true, true, false><<<gemmBlocks, 256, 0, stream>>>(
        hagg, eagg, W1, b1, inv, hA, hB);

    // ---- layer 2: no residual, no relu, hB -> out (NT stores) --------------
    (void)hipMemsetAsync(hagg, 0, ND * sizeof(float), stream);
    gcn_scatter_h_kernel<<<scatterBlocks, 256, 0, stream>>>(hB, src, dst, hagg);
    gcn_gemm_kernel<false, false, true><<<gemmBlocks, 256, 0, stream>>>(
        hagg, eagg, W2, b2, inv, nullptr, out);
}